// NeuralSemiLagrangian_48979807043787
// MI455X (gfx1250) — compile-verified
//
#include <hip/hip_runtime.h>
#include <hip/hip_bf16.h>

// ---------------------------------------------------------------------------
// NeuralSemiLagrangian for MI455X (gfx1250, wave32, WMMA)
// B=4, C=384, V=8, H=181, W=360, PAD=2, bicubic A=-0.75
// ---------------------------------------------------------------------------

#define B_   4
#define C_   384
#define V_   8
#define H_   181
#define W_   360
#define HW_  (H_ * W_)
#define NPIX (B_ * HW_)          // 260640, divisible by 16
#define PAD_ 2
#define ALPHA_ (-0.75f)

typedef float v2f __attribute__((ext_vector_type(2)));
typedef float v8f __attribute__((ext_vector_type(8)));

// workspace layout in floats
#define WS_MEAN  0
#define WS_RSTD  (WS_MEAN + NPIX)
#define WS_PROJ  (WS_RSTD + NPIX)                 // B*V*H*W
#define WS_COORD (WS_PROJ + B_ * V_ * HW_)        // 2 * B*V*H*W (x,y interleaved)

// ---------------------------------------------------------------------------
// K1: per-pixel layernorm statistics (mean, rstd) over C
// ---------------------------------------------------------------------------
__global__ void __launch_bounds__(256) ln_stats_kernel(
    const float* __restrict__ x, float* __restrict__ mean, float* __restrict__ rstd)
{
    int p = blockIdx.x * 256 + threadIdx.x;
    if (p >= NPIX) return;
    int b = p / HW_, hw = p % HW_;
    const float* base = x + (size_t)b * C_ * HW_ + hw;
    float s = 0.f, s2 = 0.f;
    #pragma unroll 4
    for (int c = 0; c < C_; ++c) {
        float v = base[c * HW_];
        s += v; s2 += v * v;
    }
    float m = s * (1.0f / C_);
    float var = s2 * (1.0f / C_) - m * m;
    mean[p] = m;
    rstd[p] = rsqrtf(var + 1e-5f);
}

// geo-cyclic pad(1) neighbor: returns spatial index (row*W + col)
__device__ __forceinline__ int geo_nb_index(int h, int w, int dh, int dw)
{
    int col = w + dw;
    if (col < 0)    col += W_;
    if (col >= W_)  col -= W_;
    int hh = h + dh;
    int row;
    if (hh < 0)        { row = 0;      col = (col + W_ / 2) % W_; }
    else if (hh >= H_) { row = H_ - 1; col = (col + W_ / 2) % W_; }
    else               { row = hh; }
    return row * W_ + col;
}

// ---------------------------------------------------------------------------
// K2: fused LN-apply + depthwise3x3 + (pw 384->16) + (down 384->8) GEMMs via
// V_WMMA_F32_16X16X4_F32, then spherical-trig departure coordinates.
// One wave32 handles a tile of 16 pixels (M); N=16 outputs; K=384 in 96x4.
// ---------------------------------------------------------------------------
__global__ void __launch_bounds__(128) velproj_kernel(
    const float* __restrict__ x,
    const float* __restrict__ mean, const float* __restrict__ rstd,
    const float* __restrict__ nscale, const float* __restrict__ nbias,
    const float* __restrict__ dwk, const float* __restrict__ dwb,
    const float* __restrict__ pww, const float* __restrict__ pwb,
    const float* __restrict__ dnw, const float* __restrict__ dnb,
    const float* __restrict__ latg, const float* __restrict__ lng,
    const float* __restrict__ dtp,
    float* __restrict__ proj, float* __restrict__ coord)
{
    __shared__ float S[4][256];          // per-wave 16x16 velocity staging
    const int wave = threadIdx.x >> 5;
    const int lane = threadIdx.x & 31;
    const int tile = blockIdx.x * 4 + wave;
    const int NTILE = NPIX / 16;
    if (tile >= NTILE) return;           // wave-uniform: EXEC stays all-ones

    const int m     = lane & 15;         // A-matrix row (pixel in tile)
    const int khalf = lane >> 4;         // 0: K{0,1}, 1: K{2,3}
    const int n     = lane & 15;         // B-matrix col / output channel

    const int P  = tile * 16 + m;
    const int b  = P / HW_;
    const int hw = P % HW_;
    const int h  = hw / W_, w = hw % W_;

    // precompute 9-tap geo neighborhood + per-neighbor LN stats
    int   sidx[9];
    float nm[9], nr[9];
    #pragma unroll
    for (int t = 0; t < 9; ++t) {
        int sp = geo_nb_index(h, w, t / 3 - 1, t % 3 - 1);
        sidx[t] = sp;
        nm[t] = mean[b * HW_ + sp];
        nr[t] = rstd[b * HW_ + sp];
    }
    const size_t xbase = (size_t)b * C_ * HW_;

    v8f accv = {0.f, 0.f, 0.f, 0.f, 0.f, 0.f, 0.f, 0.f};
    v8f accp = {0.f, 0.f, 0.f, 0.f, 0.f, 0.f, 0.f, 0.f};

    for (int c0 = 0; c0 < C_; c0 += 4) {
        v2f a1, a2, bpw, bdn;
        #pragma unroll
        for (int q = 0; q < 2; ++q) {
            const int c = c0 + khalf * 2 + q;
            const float sc = nscale[c], bi = nbias[c];
            const float* kk = dwk + c * 9;
            const float* xp = x + xbase + (size_t)c * HW_;
            float acc = dwb[c];
            float center = 0.f;
            #pragma unroll
            for (int t = 0; t < 9; ++t) {
                float xv = xp[sidx[t]];
                if (t == 4) center = xv;
                float hn = (xv - nm[t]) * nr[t] * sc + bi;     // layernorm
                acc = fmaf(kk[t], hn, acc);                    // depthwise tap
            }
            a1[q] = acc;                                       // dw3x3(hn)
            a2[q] = center;                                    // raw hidden
            bpw[q] = pww[n * C_ + c];
            bdn[q] = (n < V_) ? dnw[n * C_ + c] : 0.f;
        }
        // D = A(16x4,f32) x B(4x16,f32) + C(16x16,f32)
        accv = __builtin_amdgcn_wmma_f32_16x16x4_f32(
                   false, a1, false, bpw, (short)0, accv, false, false);
        accp = __builtin_amdgcn_wmma_f32_16x16x4_f32(
                   false, a2, false, bdn, (short)0, accp, false, false);
    }

    // stage velocity (+bias) in LDS; store proj (+bias) to workspace
    float* Sv = S[wave];
    const float pb = pwb[n];
    #pragma unroll
    for (int r = 0; r < 8; ++r) {
        int M = r + 8 * khalf;                 // C/D layout: lane half selects M+8
        Sv[M * 16 + n] = accv[r] + pb;
    }
    if (n < V_) {
        const float dbias = dnb[n];
        #pragma unroll
        for (int r = 0; r < 8; ++r) {
            int M = r + 8 * khalf;
            int PP = tile * 16 + M;
            int bb = PP / HW_, hw2 = PP % HW_;
            proj[(bb * V_ + n) * HW_ + hw2] = accp[r] + dbias;
        }
    }
    asm volatile("" ::: "memory");             // order LDS read-after-write (single wave, DS in-order)

    // departure-point trig: 16 pixels x 8 velocity channels = 128 pairs
    const float dt = dtp[0];
    const float min_lat = latg[(H_ - 1) * W_], max_lat = latg[0];
    const float min_lon = lng[0],              max_lon = lng[W_ - 1];
    const float TWO_PI = 6.28318530717958647692f;
    #pragma unroll
    for (int it = 0; it < 4; ++it) {
        int pair = lane + 32 * it;
        int mm = pair & 15, vch = pair >> 4;
        float u  = Sv[mm * 16 + vch];
        float vv = Sv[mm * 16 + vch + 8];
        int PP = tile * 16 + mm;
        int bb = PP / HW_, hw2 = PP % HW_;
        float lat_p = latg[hw2];
        float lon_p = lng[hw2];
        float lon_pr = -u  * dt;
        float lat_pr = -vv * dt;
        float slp = sinf(lat_pr), clp = cosf(lat_pr);
        float sop = sinf(lon_pr), cop = cosf(lon_pr);
        float sp_ = sinf(lat_p),  cp_ = cosf(lat_p);
        float sin_lat = slp * cp_ + clp * cop * sp_;
        sin_lat = fminf(fmaxf(sin_lat, -1.f + 1e-7f), 1.f - 1e-7f);
        float lat_dep = asinf(sin_lat);
        float num = clp * sop;
        float den = clp * cop * cp_ - slp * sp_;
        float lon_dep = fmodf(lon_p + atan2f(num, den) + TWO_PI, TWO_PI);
        float pix_x = (lon_dep - min_lon) / (max_lon - min_lon) * (float)(W_ - 1);
        float pix_y = (lat_dep - min_lat) / (max_lat - min_lat) * (float)(H_ - 1);
        int ci = ((bb * V_ + vch) * HW_ + hw2) * 2;
        coord[ci]     = pix_x + (float)PAD_;   // coords in geo-padded image space
        coord[ci + 1] = pix_y + (float)PAD_;
    }
}

// bicubic weights, A=-0.75 (matches reference cubic_weights)
__device__ __forceinline__ void cubic_w(float t, float wv[4])
{
    const float A = ALPHA_;
    float t1 = t + 1.f, t2 = 1.f - t, t3 = 2.f - t;
    wv[0] = ((A * t1 - 5.f * A) * t1 + 8.f * A) * t1 - 4.f * A;
    wv[1] = ((A + 2.f) * t - (A + 3.f)) * t * t + 1.f;
    wv[2] = ((A + 2.f) * t2 - (A + 3.f)) * t2 * t2 + 1.f;
    wv[3] = ((A * t3 - 5.f * A) * t3 + 8.f * A) * t3 - 4.f * A;
}

// ---------------------------------------------------------------------------
// K3: bicubic sample of geo-padded proj (index mapping, no materialized pad)
//     fused with up-projection 8 -> 384 (weights in LDS, coalesced stores)
// ---------------------------------------------------------------------------
__global__ void __launch_bounds__(256) sample_up_kernel(
    const float* __restrict__ proj, const float* __restrict__ coord,
    const float* __restrict__ upw,  const float* __restrict__ upb,
    float* __restrict__ out)
{
    __shared__ float Wsh[C_ * V_];
    __shared__ float Bsh[C_];
    for (int i = threadIdx.x; i < C_ * V_; i += 256) Wsh[i] = upw[i];
    for (int i = threadIdx.x; i < C_;      i += 256) Bsh[i] = upb[i];
    __syncthreads();

    int p = blockIdx.x * 256 + threadIdx.x;
    if (p >= NPIX) return;
    int b = p / HW_, hw = p % HW_;

    float s[V_];
    #pragma unroll
    for (int v = 0; v < V_; ++v) {
        int ci = ((b * V_ + v) * HW_ + hw) * 2;
        float xx = coord[ci], yy = coord[ci + 1];      // padded-image coords
        float x0 = floorf(xx), y0 = floorf(yy);
        float tx = xx - x0, ty = yy - y0;
        float wx[4], wy[4];
        cubic_w(tx, wx);
        cubic_w(ty, wy);
        int ix0 = (int)x0, iy0 = (int)y0;
        const float* pbuf = proj + (b * V_ + v) * HW_;
        float acc = 0.f;
        #pragma unroll
        for (int i = 0; i < 4; ++i) {
            int yi = iy0 + i - 1;
            yi = yi < 0 ? 0 : (yi > H_ + 3 ? H_ + 3 : yi);  // clip to padded H
            int row, rsh;
            if (yi < 2)            { row = 1 - yi;          rsh = 1; }  // top flip+roll
            else if (yi >= H_ + 2) { row = 2 * H_ + 1 - yi; rsh = 1; }  // bottom flip+roll
            else                   { row = yi - 2;          rsh = 0; }
            float rowacc = 0.f;
            #pragma unroll
            for (int j = 0; j < 4; ++j) {
                int xi = ix0 + j - 1;
                xi = xi < 0 ? 0 : (xi > W_ + 3 ? W_ + 3 : xi);  // clip to padded W
                int col = (xi - 2 + W_) % W_;                   // cyclic lon wrap
                if (rsh) col = (col + W_ / 2) % W_;             // 180-deg roll on pole rows
                rowacc = fmaf(wx[j], pbuf[row * W_ + col], rowacc);
            }
            acc = fmaf(wy[i], rowacc, acc);
        }
        s[v] = acc;
    }

    // up-projection: out[b, co, h, w] = sum_v s[v] * up_w[co, v] + up_b[co]
    float* ob = out + (size_t)b * C_ * HW_ + hw;
    for (int co = 0; co < C_; ++co) {
        float a = Bsh[co];
        const float* wr = Wsh + co * V_;
        #pragma unroll
        for (int v = 0; v < V_; ++v) a = fmaf(wr[v], s[v], a);
        ob[(size_t)co * HW_] = a;
    }
}

// ---------------------------------------------------------------------------
extern "C" void kernel_launch(void* const* d_in, const int* in_sizes, int n_in,
                              void* d_out, int out_size, void* d_ws, size_t ws_size,
                              hipStream_t stream)
{
    (void)in_sizes; (void)n_in; (void)out_size; (void)ws_size;
    const float* hidden = (const float*)d_in[0];
    const float* latg   = (const float*)d_in[1];
    const float* lng    = (const float*)d_in[2];
    const float* dtp    = (const float*)d_in[3];
    const float* nsc    = (const float*)d_in[4];
    const float* nbi    = (const float*)d_in[5];
    const float* dwk    = (const float*)d_in[6];
    const float* dwb    = (const float*)d_in[7];
    const float* pww    = (const float*)d_in[8];
    const float* pwb    = (const float*)d_in[9];
    const float* dnw    = (const float*)d_in[10];
    const float* dnb    = (const float*)d_in[11];
    const float* upw    = (const float*)d_in[12];
    const float* upb    = (const float*)d_in[13];

    float* ws    = (float*)d_ws;
    float* mean  = ws + WS_MEAN;
    float* rstd  = ws + WS_RSTD;
    float* proj  = ws + WS_PROJ;
    float* coord = ws + WS_COORD;
    float* out   = (float*)d_out;

    const int pblocks = (NPIX + 255) / 256;
    hipLaunchKernelGGL(ln_stats_kernel, dim3(pblocks), dim3(256), 0, stream,
                       hidden, mean, rstd);

    const int ntile = NPIX / 16;
    const int vblocks = (ntile + 3) / 4;           // 4 waves (tiles) per block
    hipLaunchKernelGGL(velproj_kernel, dim3(vblocks), dim3(128), 0, stream,
                       hidden, mean, rstd, nsc, nbi, dwk, dwb,
                       pww, pwb, dnw, dnb, latg, lng, dtp, proj, coord);

    hipLaunchKernelGGL(sample_up_kernel, dim3(pblocks), dim3(256), 0, stream,
                       proj, coord, upw, upb, out);
}